// SpectralCutRegularizer_59158879535699
// MI455X (gfx1250) — compile-verified
//
#include <hip/hip_runtime.h>
#include <hip/hip_bf16.h>

typedef __attribute__((ext_vector_type(2))) float v2f;
typedef __attribute__((ext_vector_type(8))) float v8f;

#define EPSF 1e-8f
constexpr int D   = 128;   // feature dim (fixed by reference)
constexpr int KC  = 32;    // K-chunk staged in LDS per iteration
constexpr int HPAD = 2;    // LDS row pad for the transposed H tile

// ---------------------------------------------------------------------------
// Kernel 1: diagonals -> dv^{-1/2}, de^{-1}; zero the two accumulators.
// ---------------------------------------------------------------------------
__global__ void prep_kernel(const float* __restrict__ Dv,
                            const float* __restrict__ De,
                            float* __restrict__ dvis,
                            float* __restrict__ deinv,
                            float* __restrict__ accs,
                            int N, int E) {
    int i = blockIdx.x * blockDim.x + threadIdx.x;
    if (i < N) dvis[i]  = rsqrtf(Dv[(size_t)i * (size_t)(N + 1)] + EPSF);
    if (i < E) deinv[i] = 1.0f / (De[(size_t)i * (size_t)(E + 1)] + EPSF);
    if (i == 0) { accs[0] = 0.0f; accs[1] = 0.0f; }
}

// ---------------------------------------------------------------------------
// Kernel 2: Fs = F * dv^{-1/2} (row scale), fused with sum(F*F) reduction.
// ---------------------------------------------------------------------------
__global__ void fs_kernel(const float* __restrict__ F,
                          const float* __restrict__ dvis,
                          float* __restrict__ Fs,
                          float* __restrict__ accs,
                          int total) {               // total = N*D
    int idx = blockIdx.x * blockDim.x + threadIdx.x;
    float v = 0.0f;
    if (idx < total) {
        float f = F[idx];
        Fs[idx] = f * dvis[idx >> 7];               // row = idx / 128
        v = f * f;
    }
    #pragma unroll
    for (int off = 16; off > 0; off >>= 1)
        v += __shfl_xor(v, off, 32);
    if ((threadIdx.x & 31) == 0)
        atomicAdd(&accs[0], v);
}

// ---------------------------------------------------------------------------
// Kernel 3: quad = sum_e de_inv[e] * || (H^T Fs)[e,:] ||^2 via FP32 WMMA.
// One block per 16-row e-tile; 8 wave32s cover the 128 d-columns, so the
// 200 MB H matrix is streamed from HBM exactly once. K double-buffered in LDS.
// ---------------------------------------------------------------------------
__global__ void __launch_bounds__(256, 2)
gemm_kernel(const float* __restrict__ H,
            const float* __restrict__ Fs,
            const float* __restrict__ deinv,
            float* __restrict__ accs,
            int N, int E) {
    __shared__ float sh[2][16][KC + HPAD];  // H tile, transposed: [e_local][k]
    __shared__ float sf[2][KC][D];          // Fs tile: [k][d]

    const int t     = threadIdx.x;
    const int wave  = t >> 5;
    const int lane  = t & 31;
    const int ln    = lane & 15;
    const int half  = lane >> 4;
    const int eBase = blockIdx.x * 16;
    const int dCol  = wave * 16 + ln;       // this wave's d-column for B/C

    v8f c = {};

    auto stage = [&](int buf, int k0) {
        // H[k0..k0+31, eBase..eBase+15] -> sh (transposed), zero-padded OOB
        #pragma unroll
        for (int r = 0; r < 2; ++r) {
            int idx = t + 256 * r;          // 0..511
            int kk = idx >> 4, m = idx & 15;
            int gi = k0 + kk, ge = eBase + m;
            float v = 0.0f;
            if (gi < N && ge < E) v = H[(size_t)gi * (size_t)E + (size_t)ge];
            sh[buf][m][kk] = v;
        }
        // Fs[k0..k0+31, 0..127] -> sf, fully coalesced
        #pragma unroll
        for (int r = 0; r < 16; ++r) {
            int idx = t + 256 * r;          // 0..4095
            int kk = idx >> 7, dc = idx & 127;
            int gi = k0 + kk;
            sf[buf][kk][dc] = (gi < N) ? Fs[(size_t)gi * D + dc] : 0.0f;
        }
    };

    stage(0, 0);
    __syncthreads();

    const int nChunks = (N + KC - 1) / KC;
    for (int ch = 0; ch < nChunks; ++ch) {
        const int buf = ch & 1;
        if (ch + 1 < nChunks) stage(buf ^ 1, (ch + 1) * KC);

        #pragma unroll
        for (int kk = 0; kk < KC; kk += 4) {
            // ISA 32-bit A 16x4 layout: VGPR0 = K(kk+2*half), VGPR1 = next K.
            const int kA = kk + 2 * half;
            v2f a, b;
            a.x = sh[buf][ln][kA];
            a.y = sh[buf][ln][kA + 1];
            b.x = sf[buf][kA][dCol];
            b.y = sf[buf][kA + 1][dCol];
            c = __builtin_amdgcn_wmma_f32_16x16x4_f32(
                    false, a, false, b, (short)0, c, false, false);
        }
        __syncthreads();
    }

    // C layout: VGPR v, lane-half -> row M = v + 8*half; col N = lane&15.
    float part = 0.0f;
    #pragma unroll
    for (int v = 0; v < 8; ++v) {
        int e = eBase + v + 8 * half;
        float de = (e < E) ? deinv[e] : 0.0f;
        float m = c[v];
        part += de * m * m;
    }
    #pragma unroll
    for (int off = 16; off > 0; off >>= 1)
        part += __shfl_xor(part, off, 32);
    if (lane == 0)
        atomicAdd(&accs[1], part);
}

// ---------------------------------------------------------------------------
// Kernel 4: loss = (sumF2 - quad) / N
// ---------------------------------------------------------------------------
__global__ void final_kernel(const float* __restrict__ accs,
                             float* __restrict__ out,
                             float invN) {
    if (blockIdx.x == 0 && threadIdx.x == 0)
        out[0] = (accs[0] - accs[1]) * invN;
}

// ---------------------------------------------------------------------------
extern "C" void kernel_launch(void* const* d_in, const int* in_sizes, int n_in,
                              void* d_out, int out_size, void* d_ws, size_t ws_size,
                              hipStream_t stream) {
    const float* F  = (const float*)d_in[0];   // [N, 128]
    const float* H  = (const float*)d_in[1];   // [N, E]
    const float* Dv = (const float*)d_in[2];   // [N, N] (diag only)
    const float* De = (const float*)d_in[3];   // [E, E] (diag only)
    float* out = (float*)d_out;

    const int N = in_sizes[0] / D;             // 10000
    const int E = in_sizes[1] / N;             // 5000

    // Workspace layout (floats): dvis[N] | deinv[E] | accs[16 pad] | Fs[N*D]
    float* ws    = (float*)d_ws;
    float* dvis  = ws;
    float* deinv = ws + N;
    float* accs  = ws + N + E;
    float* Fs    = ws + N + E + 16;

    const int threads = 256;
    const int mx = (N > E) ? N : E;

    prep_kernel<<<(mx + threads - 1) / threads, threads, 0, stream>>>(
        Dv, De, dvis, deinv, accs, N, E);

    const int total = N * D;
    fs_kernel<<<(total + threads - 1) / threads, threads, 0, stream>>>(
        F, dvis, Fs, accs, total);

    gemm_kernel<<<(E + 15) / 16, threads, 0, stream>>>(
        H, Fs, deinv, accs, N, E);

    final_kernel<<<1, 64, 0, stream>>>(accs, out, 1.0f / (float)N);
}